// CudaRendererGpu_69879117906798
// MI455X (gfx1250) — compile-verified
//
#include <hip/hip_runtime.h>
#include <math.h>

// Problem constants (match reference setup_inputs)
#define BB 2
#define CC 4
#define HH 512
#define WW 512
#define VV 100000
#define FF 200000
#define AA 8
#define HWPIX (HH * WW)

typedef __attribute__((ext_vector_type(2))) float v2f;
typedef __attribute__((ext_vector_type(8))) float v8f;

// ---------------------------------------------------------------------------
// Kernel 1: per-(b,f) face normals: cross(v1-v0, v2-v0)
// ---------------------------------------------------------------------------
__global__ __launch_bounds__(256) void face_normal_kernel(
    const float* __restrict__ vpos,   // (B,V,3)
    const int*   __restrict__ faces,  // (F,3)
    float*       __restrict__ fn)     // (B,F,3)
{
    int idx = blockIdx.x * blockDim.x + threadIdx.x;
    const int total = BB * FF;
    if (idx >= total) return;
    int b = idx / FF;
    int f = idx - b * FF;

    int i0 = faces[f * 3 + 0];
    int i1 = faces[f * 3 + 1];
    int i2 = faces[f * 3 + 2];
    const float* vb = vpos + (size_t)b * VV * 3;

    float ax = vb[i0 * 3 + 0], ay = vb[i0 * 3 + 1], az = vb[i0 * 3 + 2];
    float bx = vb[i1 * 3 + 0], by = vb[i1 * 3 + 1], bz = vb[i1 * 3 + 2];
    float cx = vb[i2 * 3 + 0], cy = vb[i2 * 3 + 1], cz = vb[i2 * 3 + 2];

    float e1x = bx - ax, e1y = by - ay, e1z = bz - az;
    float e2x = cx - ax, e2y = cy - ay, e2z = cz - az;

    float nx = e1y * e2z - e1z * e2y;
    float ny = e1z * e2x - e1x * e2z;
    float nz = e1x * e2y - e1y * e2x;

    fn[(size_t)idx * 3 + 0] = nx;
    fn[(size_t)idx * 3 + 1] = ny;
    fn[(size_t)idx * 3 + 2] = nz;
}

// ---------------------------------------------------------------------------
// Kernel 2: per-(b,v) vertex normals: weighted sum over A adjacent faces
// ---------------------------------------------------------------------------
__global__ __launch_bounds__(256) void vertex_normal_kernel(
    const int*   __restrict__ adj,   // (V,A)
    const float* __restrict__ wts,   // (V,A)
    const float* __restrict__ fn,    // (B,F,3)
    float*       __restrict__ vn)    // (B,V,3)
{
    int idx = blockIdx.x * blockDim.x + threadIdx.x;
    const int total = BB * VV;
    if (idx >= total) return;
    int b = idx / VV;
    int v = idx - b * VV;

    float nx = 0.f, ny = 0.f, nz = 0.f;
#pragma unroll
    for (int a = 0; a < AA; ++a) {
        int   fi = adj[v * AA + a];
        float w  = wts[v * AA + a];
        const float* p = fn + ((size_t)b * FF + fi) * 3;
        nx += w * p[0];
        ny += w * p[1];
        nz += w * p[2];
    }
    vn[(size_t)idx * 3 + 0] = nx;
    vn[(size_t)idx * 3 + 1] = ny;
    vn[(size_t)idx * 3 + 2] = nz;
}

// ---------------------------------------------------------------------------
// Kernel 3: per-camera screen transform via V_WMMA_F32_16X16X4_F32.
//   M = intrinsics(3x3) @ extrinsics(3x4); per tile of 16 vertices:
//   A(16x4) = [x y z 1] rows, B(4x16) = M^T zero-padded, D(16x16) -> cols 0..2
//   iv = [p0/p2, p1/p2, p2]
// A layout (ISA 7.12.2, 32-bit A 16x4): lanes 0-15 hold K=0 (V0) / K=1 (V1)
// for row M=lane; lanes 16-31 hold K=2 / K=3 for row M=lane-16.
// B layout (row-split rule, 4 rows over 2 VGPRs): V0 = K0 (lanes 0-15),
// K2 (lanes 16-31); V1 = K1, K3.  D: VGPR r = row r (lanes 0-15, col=lane),
// row r+8 (lanes 16-31).
// ---------------------------------------------------------------------------
__global__ __launch_bounds__(256) void screen_xform_kernel(
    const float* __restrict__ vpos,  // (B,V,3)
    const float* __restrict__ intr,  // (B,C,3,3)
    const float* __restrict__ extr,  // (B,C,3,4)
    float*       __restrict__ iv)    // (B,C,V,3)
{
    const int cam = blockIdx.y;      // b*C + c  (uniform per block)
    const int b   = cam / CC;

    // Combined camera matrix M = K @ E (3x4); uniform -> scalarized loads.
    const float* K = intr + cam * 9;
    const float* E = extr + cam * 12;
    float M[3][4];
#pragma unroll
    for (int j = 0; j < 3; ++j)
#pragma unroll
        for (int k = 0; k < 4; ++k)
            M[j][k] = K[j * 3 + 0] * E[0 * 4 + k]
                    + K[j * 3 + 1] * E[1 * 4 + k]
                    + K[j * 3 + 2] * E[2 * 4 + k];

    const int lane  = threadIdx.x & 31;
    const int wave  = threadIdx.x >> 5;
    const int wpb   = blockDim.x >> 5;
    const int n     = lane & 15;
    const bool hi   = lane >= 16;

    const float* vb  = vpos + (size_t)b * VV * 3;
    float*       ivc = iv   + (size_t)cam * VV * 3;

    const int numTiles = (VV + 15) / 16;

#if __has_builtin(__builtin_amdgcn_wmma_f32_16x16x4_f32)
    // B operand is loop-invariant.
    float b0 = 0.f, b1 = 0.f;
    if (n < 3) {
        b0 = hi ? M[n][2] : M[n][0];
        b1 = hi ? M[n][3] : M[n][1];
    }
    v2f Bv; Bv.x = b0; Bv.y = b1;
    const int srcByte = (((lane & 16) | 2) << 2);  // ds_bpermute byte index of w-col

    for (int tile = blockIdx.x * wpb + wave; tile < numTiles;
         tile += gridDim.x * wpb) {
        const int base = tile * 16;
        const int vA   = base + n;
        float a0 = 0.f, a1 = 0.f;
        if (vA < VV) {
            if (!hi) { a0 = vb[vA * 3 + 0]; a1 = vb[vA * 3 + 1]; }
            else     { a0 = vb[vA * 3 + 2]; a1 = 1.0f; }
        }
        v2f Av; Av.x = a0; Av.y = a1;
        v8f Cz = {0.f, 0.f, 0.f, 0.f, 0.f, 0.f, 0.f, 0.f};

        // EXEC is all-ones here (loop condition is wave-uniform).
        v8f D = __builtin_amdgcn_wmma_f32_16x16x4_f32(
            false, Av, false, Bv, (short)0, Cz, false, false);

#pragma unroll
        for (int r = 0; r < 8; ++r) {
            float val = D[r];
            // Broadcast this row's w (col 2) from lane 2 / lane 18.
            float w = __int_as_float(
                __builtin_amdgcn_ds_bpermute(srcByte, __float_as_int(val)));
            int m  = r + (hi ? 8 : 0);
            int vv = base + m;
            if (n < 3 && vv < VV) {
                float o = (n < 2) ? (val / w) : val;
                ivc[(size_t)vv * 3 + n] = o;
            }
        }
    }
#else
    // Fallback (also what the host pass parses): direct per-lane transform.
    for (int tile = blockIdx.x * wpb + wave; tile < numTiles;
         tile += gridDim.x * wpb) {
        const int base = tile * 16;
        const int vA   = base + n;
        if (!hi && vA < VV) {
            float x = vb[vA * 3 + 0], y = vb[vA * 3 + 1], z = vb[vA * 3 + 2];
            float p0 = M[0][0] * x + M[0][1] * y + M[0][2] * z + M[0][3];
            float p1 = M[1][0] * x + M[1][1] * y + M[1][2] * z + M[1][3];
            float p2 = M[2][0] * x + M[2][1] * y + M[2][2] * z + M[2][3];
            ivc[(size_t)vA * 3 + 0] = p0 / p2;
            ivc[(size_t)vA * 3 + 1] = p1 / p2;
            ivc[(size_t)vA * 3 + 2] = p2;
        }
    }
#endif
}

// ---------------------------------------------------------------------------
// Kernel 4: fused per-pixel pass (the bandwidth-dominant stage).
// grid.y = cam so camera params are block-uniform (scalar loads).
// ---------------------------------------------------------------------------
__global__ __launch_bounds__(256) void pixel_kernel(
    const float* __restrict__ vpos,      // (B,V,3)
    const int*   __restrict__ faces,     // (F,3)
    const float* __restrict__ inv_extr,  // (B,C,4,4)
    const int*   __restrict__ face_id,   // (B,C,H,W)
    const float* __restrict__ bary,      // (B,C,H,W,3)
    const float* __restrict__ vn,        // (B,V,3)
    const float* __restrict__ iv,        // (B,C,V,3)
    float*       __restrict__ out)       // (5,B,C,H,W,3)
{
    const int cam = blockIdx.y;
    const int b   = cam / CC;
    const int pix = blockIdx.x * blockDim.x + threadIdx.x;
    if (pix >= HWPIX) return;
    const size_t pidx = (size_t)cam * HWPIX + pix;

    // Stream prefetch of the next block's barycentrics (global_prefetch_b8;
    // speculative — safely dropped past the end of the buffer).
    __builtin_prefetch(&bary[(pidx + 256) * 3], 0, 1);

    const int  fid  = face_id[pidx];
    const bool mask = fid >= 0;
    const int  f    = mask ? fid : 0;

    const int i0 = faces[f * 3 + 0];
    const int i1 = faces[f * 3 + 1];
    const int i2 = faces[f * 3 + 2];

    const float w0 = bary[pidx * 3 + 0];
    const float w1 = bary[pidx * 3 + 1];
    const float w2 = bary[pidx * 3 + 2];

    const float* vb = vpos + (size_t)b * VV * 3;
    const float* nb = vn   + (size_t)b * VV * 3;
    const float* sb = iv   + (size_t)cam * VV * 3;

    float px = w0 * vb[i0 * 3 + 0] + w1 * vb[i1 * 3 + 0] + w2 * vb[i2 * 3 + 0];
    float py = w0 * vb[i0 * 3 + 1] + w1 * vb[i1 * 3 + 1] + w2 * vb[i2 * 3 + 1];
    float pz = w0 * vb[i0 * 3 + 2] + w1 * vb[i1 * 3 + 2] + w2 * vb[i2 * 3 + 2];

    float nx = w0 * nb[i0 * 3 + 0] + w1 * nb[i1 * 3 + 0] + w2 * nb[i2 * 3 + 0];
    float ny = w0 * nb[i0 * 3 + 1] + w1 * nb[i1 * 3 + 1] + w2 * nb[i2 * 3 + 1];
    float nz = w0 * nb[i0 * 3 + 2] + w1 * nb[i1 * 3 + 2] + w2 * nb[i2 * 3 + 2];

    float sx = w0 * sb[i0 * 3 + 0] + w1 * sb[i1 * 3 + 0] + w2 * sb[i2 * 3 + 0];
    float sy = w0 * sb[i0 * 3 + 1] + w1 * sb[i1 * 3 + 1] + w2 * sb[i2 * 3 + 1];
    float sz = w0 * sb[i0 * 3 + 2] + w1 * sb[i1 * 3 + 2] + w2 * sb[i2 * 3 + 2];

    // Normalize blended normal (clip(|n|, 1e-12))
    float nlen  = sqrtf(nx * nx + ny * ny + nz * nz);
    float ninv  = 1.0f / fmaxf(nlen, 1e-12f);
    float ux = nx * ninv, uy = ny * ninv, uz = nz * ninv;

    // Camera origin from inverse_extrinsics column 3 (block-uniform).
    const float* ie = inv_extr + cam * 16;
    float iw = ie[15];
    float ox = ie[3] / iw, oy = ie[7] / iw, oz = ie[11] / iw;

    float dx = ox - px, dy = oy - py, dz = oz - pz;
    float depth = sqrtf(dx * dx + dy * dy + dz * dz);

    const size_t istride = (size_t)BB * CC * HWPIX * 3;
    const size_t o       = pidx * 3;
    const float  fg      = mask ? 1.0f : 0.0f;

    out[0 * istride + o + 0] = fg;
    out[0 * istride + o + 1] = fg;
    out[0 * istride + o + 2] = fg;

    out[1 * istride + o + 0] = mask ? px : 0.0f;
    out[1 * istride + o + 1] = mask ? py : 0.0f;
    out[1 * istride + o + 2] = mask ? pz : 0.0f;

    out[2 * istride + o + 0] = mask ? ux : 0.0f;
    out[2 * istride + o + 1] = mask ? uy : 0.0f;
    out[2 * istride + o + 2] = mask ? uz : 0.0f;

    out[3 * istride + o + 0] = mask ? depth : 0.0f;
    out[3 * istride + o + 1] = mask ? depth : 0.0f;
    out[3 * istride + o + 2] = mask ? depth : 0.0f;

    out[4 * istride + o + 0] = mask ? sx : 0.0f;
    out[4 * istride + o + 1] = mask ? sy : 0.0f;
    out[4 * istride + o + 2] = mask ? sz : 0.0f;
}

// ---------------------------------------------------------------------------
extern "C" void kernel_launch(void* const* d_in, const int* in_sizes, int n_in,
                              void* d_out, int out_size, void* d_ws, size_t ws_size,
                              hipStream_t stream) {
    const float* vertex_pos         = (const float*)d_in[0];
    const int*   faces              = (const int*)  d_in[1];
    const int*   vert_adj_faces     = (const int*)  d_in[2];
    const float* vert_adj_weights   = (const float*)d_in[3];
    const float* intrinsics         = (const float*)d_in[4];
    const float* extrinsics         = (const float*)d_in[5];
    const float* inverse_extrinsics = (const float*)d_in[6];
    const int*   face_id            = (const int*)  d_in[7];
    const float* bary               = (const float*)d_in[8];
    float*       out                = (float*)d_out;

    // Workspace layout (floats): fn (B*F*3) | vn (B*V*3) | iv (B*C*V*3) = 16.8 MB
    float* fn = (float*)d_ws;
    float* vn = fn + (size_t)BB * FF * 3;
    float* iv = vn + (size_t)BB * VV * 3;

    {
        int total = BB * FF;
        face_normal_kernel<<<(total + 255) / 256, 256, 0, stream>>>(
            vertex_pos, faces, fn);
    }
    {
        int total = BB * VV;
        vertex_normal_kernel<<<(total + 255) / 256, 256, 0, stream>>>(
            vert_adj_faces, vert_adj_weights, fn, vn);
    }
    {
        dim3 grid(128, BB * CC);
        screen_xform_kernel<<<grid, 256, 0, stream>>>(
            vertex_pos, intrinsics, extrinsics, iv);
    }
    {
        dim3 grid(HWPIX / 256, BB * CC);
        pixel_kernel<<<grid, 256, 0, stream>>>(
            vertex_pos, faces, inverse_extrinsics, face_id, bary, vn, iv, out);
    }
}